// HiResPrecipNet_TEST_3_85057532330034
// MI455X (gfx1250) — compile-verified
//
#include <hip/hip_runtime.h>
#include <hip/hip_bf16.h>

// ---------------------------------------------------------------------------
// HiResPrecipNet forward for gfx1250 (MI455X).
//  - GRU+dense encoder: 1 thread/node, GRU weights staged to LDS via
//    GLOBAL_LOAD_ASYNC_TO_LDS_B32 (ASYNCcnt-tracked CDNA5 path).
//  - All D=16 node transforms use V_WMMA_F32_16X16X4_F32 (f32 = reference dtype).
//  - Edge scatter/softmax via global f32 atomics + order-preserving uint atomicMax.
//  - BatchNorms folded into following transform as per-feature scale/shift.
// ---------------------------------------------------------------------------

typedef __attribute__((ext_vector_type(2))) float v2f;
typedef __attribute__((ext_vector_type(8))) float v8f;

#define T_SEQ 25
#define G_HIN 25
#define G_HH  25
#define D_ENC 32
#define D_HI  16
#define SLOPE 0.2f
#define BN_EPS 1e-5f

#if defined(__gfx1250__) && __has_builtin(__builtin_amdgcn_global_load_async_to_lds_b32) && \
    __has_builtin(__builtin_amdgcn_s_wait_asynccnt)
#define HAVE_ASYNC_LDS 1
#else
#define HAVE_ASYNC_LDS 0
#endif

#if HAVE_ASYNC_LDS
typedef __attribute__((address_space(1))) int gint_as;   // global dword
typedef __attribute__((address_space(3))) int lint_as;   // LDS dword
#endif

// ---------------- utility ----------------
__global__ void k_zero_f(float* __restrict__ p, int n) {
  int i = blockIdx.x * blockDim.x + threadIdx.x;
  if (i < n) p[i] = 0.0f;
}

__device__ __forceinline__ unsigned fenc(float f) {
  unsigned u = __float_as_uint(f);
  return (u & 0x80000000u) ? ~u : (u | 0x80000000u);
}
__device__ __forceinline__ float fdec(unsigned u) {
  unsigned v = (u & 0x80000000u) ? (u & 0x7FFFFFFFu) : ~u;
  return __uint_as_float(v);
}

// ---------------- GRU + dense encoder (fused) ----------------
__global__ void k_gru_dense(const float* __restrict__ x_low,
                            const float* __restrict__ Wih, const float* __restrict__ Whh,
                            const float* __restrict__ bih, const float* __restrict__ bhh,
                            const float* __restrict__ dW,  const float* __restrict__ db,
                            float* __restrict__ enc_out, float* __restrict__ stats,
                            int n_low) {
  __shared__ float sWih[75 * 25], sWhh[75 * 25], sbih[75], sbhh[75];
  __shared__ float ssum[32], ssq[32];
  int tid = threadIdx.x;
#if HAVE_ASYNC_LDS
  // CDNA5 async Global->LDS DMA path (ASYNCcnt-tracked), per CDNA5 ISA.
  for (int i = tid; i < 75 * 25; i += blockDim.x) {
    __builtin_amdgcn_global_load_async_to_lds_b32(
        (gint_as*)(Wih + i), (lint_as*)(&sWih[i]), 0, 0);
    __builtin_amdgcn_global_load_async_to_lds_b32(
        (gint_as*)(Whh + i), (lint_as*)(&sWhh[i]), 0, 0);
  }
  for (int i = tid; i < 75; i += blockDim.x) {
    __builtin_amdgcn_global_load_async_to_lds_b32(
        (gint_as*)(bih + i), (lint_as*)(&sbih[i]), 0, 0);
    __builtin_amdgcn_global_load_async_to_lds_b32(
        (gint_as*)(bhh + i), (lint_as*)(&sbhh[i]), 0, 0);
  }
  __builtin_amdgcn_s_wait_asynccnt(0);
#else
  for (int i = tid; i < 75 * 25; i += blockDim.x) { sWih[i] = Wih[i]; sWhh[i] = Whh[i]; }
  for (int i = tid; i < 75; i += blockDim.x) { sbih[i] = bih[i]; sbhh[i] = bhh[i]; }
#endif
  if (tid < 32) { ssum[tid] = 0.0f; ssq[tid] = 0.0f; }
  __syncthreads();

  int nd = blockIdx.x * blockDim.x + tid;
  if (nd < n_low) {
    float h[G_HH], r[G_HH], z[G_HH], xt[G_HIN], enc[D_ENC];
    for (int k = 0; k < G_HH; k++) h[k] = 0.0f;
    for (int j = 0; j < D_ENC; j++) enc[j] = 0.0f;
    const float* xp = x_low + (size_t)nd * (T_SEQ * G_HIN);

    for (int t = 0; t < T_SEQ; t++) {
      for (int k = 0; k < G_HIN; k++) xt[k] = xp[t * G_HIN + k];
      // r gate
      for (int g = 0; g < G_HH; g++) {
        float gi = sbih[g], gh = sbhh[g];
#pragma unroll
        for (int k = 0; k < G_HIN; k++) { gi += sWih[g * 25 + k] * xt[k]; gh += sWhh[g * 25 + k] * h[k]; }
        r[g] = 1.0f / (1.0f + __expf(-(gi + gh)));
      }
      // z gate
      for (int g = 0; g < G_HH; g++) {
        float gi = sbih[25 + g], gh = sbhh[25 + g];
#pragma unroll
        for (int k = 0; k < G_HIN; k++) { gi += sWih[(25 + g) * 25 + k] * xt[k]; gh += sWhh[(25 + g) * 25 + k] * h[k]; }
        z[g] = 1.0f / (1.0f + __expf(-(gi + gh)));
      }
      // n gate (store into r[], using old h)
      for (int g = 0; g < G_HH; g++) {
        float gi = sbih[50 + g], gh = sbhh[50 + g];
#pragma unroll
        for (int k = 0; k < G_HIN; k++) { gi += sWih[(50 + g) * 25 + k] * xt[k]; gh += sWhh[(50 + g) * 25 + k] * h[k]; }
        r[g] = tanhf(gi + r[g] * gh);
      }
      for (int g = 0; g < G_HH; g++) h[g] = (1.0f - z[g]) * r[g] + z[g] * h[g];
      // fused dense accumulation: enc += dW[:, t*25 : t*25+25] @ h
      for (int j = 0; j < D_ENC; j++) {
        float a = enc[j];
        const float* w = dW + (size_t)j * (T_SEQ * G_HH) + t * G_HH;
#pragma unroll
        for (int k = 0; k < G_HH; k++) a += w[k] * h[k];
        enc[j] = a;
      }
    }
    for (int j = 0; j < D_ENC; j++) {
      float v = enc[j] + db[j];
      v = v > 0.0f ? v : 0.0f;            // ReLU (before BN, matching reference)
      enc_out[(size_t)nd * D_ENC + j] = v;
      atomicAdd(&ssum[j], v);
      atomicAdd(&ssq[j], v * v);
    }
  }
  __syncthreads();
  if (tid < 32) { atomicAdd(&stats[tid], ssum[tid]); atomicAdd(&stats[32 + tid], ssq[tid]); }
}

// ---------------- BN finalize: s = g*rsqrt(var+eps), t = b - m*s ----------------
__global__ void k_finalize_bn(const float* __restrict__ stats,
                              const float* __restrict__ g, const float* __restrict__ b,
                              float* __restrict__ s, float* __restrict__ t,
                              int nfeat, float n) {
  int f = threadIdx.x;
  if (f < nfeat) {
    float m = stats[f] / n;
    float v = stats[nfeat + f] / n - m * m;
    float sc = g[f] * rsqrtf(v + BN_EPS);
    s[f] = sc;
    t[f] = b[f] - m * sc;
  }
}

// ---------------- WMMA: low16 = BN(enc) @ Wrel.T   [n_low x 32] x [16 x 32] ----------------
__global__ void k_down_project(const float* __restrict__ enc,
                               const float* __restrict__ s, const float* __restrict__ t,
                               const float* __restrict__ Wrel,
                               float* __restrict__ out16, int n_low) {
  int wave = (blockIdx.x * blockDim.x + threadIdx.x) >> 5;
  int lane = threadIdx.x & 31;
  int row0 = wave * 16;
  if (row0 >= n_low) return;               // wave-uniform exit: EXEC all ones for WMMA
  int half = lane >> 4, r = lane & 15, n = r;
  int m = row0 + r; if (m > n_low - 1) m = n_low - 1;
  const float* xrow = enc + (size_t)m * D_ENC;
  v8f c; for (int j = 0; j < 8; j++) c[j] = 0.0f;
#pragma unroll
  for (int kk = 0; kk < 8; kk++) {
    int k = kk * 4 + half * 2;
    v2f a; a.x = xrow[k] * s[k] + t[k]; a.y = xrow[k + 1] * s[k + 1] + t[k + 1];
    v2f bb; bb.x = Wrel[n * 32 + k];   bb.y = Wrel[n * 32 + k + 1];
    c = __builtin_amdgcn_wmma_f32_16x16x4_f32(false, a, false, bb, (short)0, c, false, false);
  }
  float* op = out16 + (size_t)(row0 + 8 * half) * D_HI + n;
  if (row0 + 16 <= n_low) {                // uniform fast path: no per-element branches
#pragma unroll
    for (int j = 0; j < 8; j++) op[(size_t)j * D_HI] = c[j];
  } else {
    for (int j = 0; j < 8; j++) {
      int mr = row0 + j + 8 * half;
      if (mr < n_low) out16[(size_t)mr * D_HI + n] = c[j];
    }
  }
}

// ---------------- WMMA: xl = act(y)@Wl.T+bl ; xr = act(y)@Wr.T+br  (act = BN[+ReLU]) ----------------
__global__ void k_gat_transform(const float* __restrict__ y,
                                const float* __restrict__ s, const float* __restrict__ t,
                                int apply_relu,
                                const float* __restrict__ Wl, const float* __restrict__ bl,
                                const float* __restrict__ Wr, const float* __restrict__ br,
                                float* __restrict__ xl, float* __restrict__ xr, int n_nodes) {
  int wave = (blockIdx.x * blockDim.x + threadIdx.x) >> 5;
  int lane = threadIdx.x & 31;
  int row0 = wave * 16;
  if (row0 >= n_nodes) return;             // wave-uniform exit
  int half = lane >> 4, r = lane & 15, n = r;
  int m = row0 + r; if (m > n_nodes - 1) m = n_nodes - 1;
  const float* xrow = y + (size_t)m * D_HI;
  v2f a[4];
#pragma unroll
  for (int kk = 0; kk < 4; kk++) {
    int k = kk * 4 + half * 2;
    float x0 = xrow[k] * s[k] + t[k];
    float x1 = xrow[k + 1] * s[k + 1] + t[k + 1];
    if (apply_relu) { x0 = fmaxf(x0, 0.0f); x1 = fmaxf(x1, 0.0f); }
    a[kk].x = x0; a[kk].y = x1;
  }
  v8f cl, cr;
  for (int j = 0; j < 8; j++) { cl[j] = bl[n]; cr[j] = br[n]; }
#pragma unroll
  for (int kk = 0; kk < 4; kk++) {
    int k = kk * 4 + half * 2;
    v2f bL; bL.x = Wl[n * 16 + k]; bL.y = Wl[n * 16 + k + 1];
    v2f bR; bR.x = Wr[n * 16 + k]; bR.y = Wr[n * 16 + k + 1];
    cl = __builtin_amdgcn_wmma_f32_16x16x4_f32(false, a[kk], false, bL, (short)0, cl, false, false);
    cr = __builtin_amdgcn_wmma_f32_16x16x4_f32(false, a[kk], false, bR, (short)0, cr, false, false);
  }
  size_t base = (size_t)(row0 + 8 * half) * D_HI + n;
  if (row0 + 16 <= n_nodes) {              // uniform fast path
#pragma unroll
    for (int j = 0; j < 8; j++) {
      xl[base + (size_t)j * D_HI] = cl[j];
      xr[base + (size_t)j * D_HI] = cr[j];
    }
  } else {
    for (int j = 0; j < 8; j++) {
      int mr = row0 + j + 8 * half;
      if (mr < n_nodes) {
        xl[(size_t)mr * D_HI + n] = cl[j];
        xr[(size_t)mr * D_HI + n] = cr[j];
      }
    }
  }
}

// ---------------- edge kernels ----------------
__global__ void k_count(const int* __restrict__ dst, float* __restrict__ cnt, int ne) {
  int e = blockIdx.x * blockDim.x + threadIdx.x;
  if (e < ne) atomicAdd(&cnt[dst[e]], 1.0f);
}

__global__ void k_scatter16(const float* __restrict__ feat, const int* __restrict__ src,
                            const int* __restrict__ dst, float* __restrict__ acc, int ne) {
  int e = blockIdx.x * blockDim.x + threadIdx.x;
  if (e >= ne) return;
  const float4* p = (const float4*)(feat + (size_t)src[e] * D_HI);
  float* a = acc + (size_t)dst[e] * D_HI;
  float4 v0 = p[0], v1 = p[1], v2 = p[2], v3 = p[3];
  atomicAdd(a + 0, v0.x);  atomicAdd(a + 1, v0.y);  atomicAdd(a + 2, v0.z);  atomicAdd(a + 3, v0.w);
  atomicAdd(a + 4, v1.x);  atomicAdd(a + 5, v1.y);  atomicAdd(a + 6, v1.z);  atomicAdd(a + 7, v1.w);
  atomicAdd(a + 8, v2.x);  atomicAdd(a + 9, v2.y);  atomicAdd(a + 10, v2.z); atomicAdd(a + 11, v2.w);
  atomicAdd(a + 12, v3.x); atomicAdd(a + 13, v3.y); atomicAdd(a + 14, v3.z); atomicAdd(a + 15, v3.w);
}

__device__ __forceinline__ void edge_ids(int idx, int e_hh,
                                         const int* __restrict__ src,
                                         const int* __restrict__ dst, int& s_, int& d_) {
  if (idx < e_hh) { s_ = src[idx]; d_ = dst[idx]; }
  else { s_ = idx - e_hh; d_ = s_; }       // appended self-loops
}

__global__ void k_edge_logit(const float* __restrict__ xl, const float* __restrict__ xr,
                             const int* __restrict__ src, const int* __restrict__ dst,
                             const float* __restrict__ att, float* __restrict__ ebuf,
                             unsigned* __restrict__ emax, int e_hh, int e_tot) {
  int idx = blockIdx.x * blockDim.x + threadIdx.x;
  if (idx >= e_tot) return;
  int s_, d_;
  edge_ids(idx, e_hh, src, dst, s_, d_);
  const float4* pl = (const float4*)(xl + (size_t)s_ * D_HI);
  const float4* pr = (const float4*)(xr + (size_t)d_ * D_HI);
  float e = 0.0f;
#pragma unroll
  for (int q = 0; q < 4; q++) {
    float4 vl = pl[q], vr = pr[q];
    float v;
    v = vl.x + vr.x; v = v > 0.0f ? v : SLOPE * v; e += att[q * 4 + 0] * v;
    v = vl.y + vr.y; v = v > 0.0f ? v : SLOPE * v; e += att[q * 4 + 1] * v;
    v = vl.z + vr.z; v = v > 0.0f ? v : SLOPE * v; e += att[q * 4 + 2] * v;
    v = vl.w + vr.w; v = v > 0.0f ? v : SLOPE * v; e += att[q * 4 + 3] * v;
  }
  ebuf[idx] = e;
  atomicMax(&emax[d_], fenc(e));
}

__global__ void k_edge_exp(const int* __restrict__ src, const int* __restrict__ dst,
                           float* __restrict__ ebuf, const unsigned* __restrict__ emax,
                           float* __restrict__ denom, int e_hh, int e_tot) {
  int idx = blockIdx.x * blockDim.x + threadIdx.x;
  if (idx >= e_tot) return;
  int s_, d_;
  edge_ids(idx, e_hh, src, dst, s_, d_);
  float ex = __expf(ebuf[idx] - fdec(emax[d_]));
  ebuf[idx] = ex;
  atomicAdd(&denom[d_], ex);
}

__global__ void k_edge_scatter(const float* __restrict__ xl,
                               const int* __restrict__ src, const int* __restrict__ dst,
                               const float* __restrict__ ebuf, const float* __restrict__ denom,
                               float* __restrict__ msg, int e_hh, int e_tot) {
  int idx = blockIdx.x * blockDim.x + threadIdx.x;
  if (idx >= e_tot) return;
  int s_, d_;
  edge_ids(idx, e_hh, src, dst, s_, d_);
  float alpha = ebuf[idx] / denom[d_];
  const float4* p = (const float4*)(xl + (size_t)s_ * D_HI);
  float* a = msg + (size_t)d_ * D_HI;
  float4 v0 = p[0], v1 = p[1], v2 = p[2], v3 = p[3];
  atomicAdd(a + 0, alpha * v0.x);  atomicAdd(a + 1, alpha * v0.y);
  atomicAdd(a + 2, alpha * v0.z);  atomicAdd(a + 3, alpha * v0.w);
  atomicAdd(a + 4, alpha * v1.x);  atomicAdd(a + 5, alpha * v1.y);
  atomicAdd(a + 6, alpha * v1.z);  atomicAdd(a + 7, alpha * v1.w);
  atomicAdd(a + 8, alpha * v2.x);  atomicAdd(a + 9, alpha * v2.y);
  atomicAdd(a + 10, alpha * v2.z); atomicAdd(a + 11, alpha * v2.w);
  atomicAdd(a + 12, alpha * v3.x); atomicAdd(a + 13, alpha * v3.y);
  atomicAdd(a + 14, alpha * v3.z); atomicAdd(a + 15, alpha * v3.w);
}

// ---------------- node kernels ----------------
__global__ void k_high_init(const float* __restrict__ agg, const float* __restrict__ cnt,
                            const float* __restrict__ z_std, const float* __restrict__ land,
                            const float* __restrict__ brel, const float* __restrict__ Wroot,
                            float* __restrict__ y0, float* __restrict__ stats, int n_high) {
  __shared__ float ssum[16], ssq[16];
  int tid = threadIdx.x;
  if (tid < 16) { ssum[tid] = 0.0f; ssq[tid] = 0.0f; }
  __syncthreads();
  int nd = blockIdx.x * blockDim.x + tid;
  if (nd < n_high) {
    float zl[7];
    for (int q = 0; q < 6; q++) zl[q] = z_std[(size_t)nd * 6 + q];
    zl[6] = land[nd];
    float c = cnt[nd]; if (c < 1.0f) c = 1.0f;
    float inv = 1.0f / c;
    for (int f = 0; f < D_HI; f++) {
      float v = agg[(size_t)nd * D_HI + f] * inv + brel[f];
#pragma unroll
      for (int q = 0; q < 7; q++) v += Wroot[f * 7 + q] * zl[q];
      y0[(size_t)nd * D_HI + f] = v;
      atomicAdd(&ssum[f], v); atomicAdd(&ssq[f], v * v);
    }
  }
  __syncthreads();
  if (tid < 16) { atomicAdd(&stats[tid], ssum[tid]); atomicAdd(&stats[16 + tid], ssq[tid]); }
}

__global__ void k_gat_finalize(const float* __restrict__ msg, const float* __restrict__ cnt,
                               const float* __restrict__ bias, float* __restrict__ yout,
                               float* __restrict__ stats, int collect, int n_high) {
  __shared__ float ssum[16], ssq[16];
  int tid = threadIdx.x;
  if (tid < 16) { ssum[tid] = 0.0f; ssq[tid] = 0.0f; }
  __syncthreads();
  int nd = blockIdx.x * blockDim.x + tid;
  if (nd < n_high) {
    float inv = 1.0f / (cnt[nd] + 1.0f);   // +1 for self-loop
    for (int f = 0; f < D_HI; f++) {
      float v = msg[(size_t)nd * D_HI + f] * inv + bias[f];
      yout[(size_t)nd * D_HI + f] = v;
      if (collect) { atomicAdd(&ssum[f], v); atomicAdd(&ssq[f], v * v); }
    }
  }
  __syncthreads();
  if (collect && tid < 16) { atomicAdd(&stats[tid], ssum[tid]); atomicAdd(&stats[16 + tid], ssq[tid]); }
}

__global__ void k_predict(const float* __restrict__ y, const float* __restrict__ pw,
                          const float* __restrict__ pb, float* __restrict__ out, int n_high) {
  int nd = blockIdx.x * blockDim.x + threadIdx.x;
  if (nd >= n_high) return;
  float acc = pb[0];
#pragma unroll
  for (int f = 0; f < D_HI; f++) {
    float v = y[(size_t)nd * D_HI + f];
    v = v > 0.0f ? v : 0.0f;               // final ReLU (layer 4, no BN)
    acc += pw[f] * v;
  }
  out[nd] = acc;
}

// ---------------- host orchestration ----------------
#define CDIV(a, b) (((a) + (b) - 1) / (b))

extern "C" void kernel_launch(void* const* d_in, const int* in_sizes, int n_in,
                              void* d_out, int out_size, void* d_ws, size_t ws_size,
                              hipStream_t stream) {
  const float* x_low   = (const float*)d_in[0];
  const float* z_std   = (const float*)d_in[1];
  const float* land    = (const float*)d_in[2];
  const int*   l2h_src = (const int*)d_in[3];
  const int*   l2h_dst = (const int*)d_in[4];
  const int*   hh_src  = (const int*)d_in[5];
  const int*   hh_dst  = (const int*)d_in[6];
  const float* gru_Wih = (const float*)d_in[7];
  const float* gru_Whh = (const float*)d_in[8];
  const float* gru_bih = (const float*)d_in[9];
  const float* gru_bhh = (const float*)d_in[10];
  const float* dense_W = (const float*)d_in[11];
  const float* dense_b = (const float*)d_in[12];
  const float* bn_enc_g = (const float*)d_in[13];
  const float* bn_enc_b = (const float*)d_in[14];
  const float* down_Wrel = (const float*)d_in[15];
  const float* down_brel = (const float*)d_in[16];
  const float* down_Wroot = (const float*)d_in[17];
  const float* gat_Wl  = (const float*)d_in[18];
  const float* gat_bl  = (const float*)d_in[19];
  const float* gat_Wr  = (const float*)d_in[20];
  const float* gat_br  = (const float*)d_in[21];
  const float* gat_att = (const float*)d_in[22];
  const float* gat_bias = (const float*)d_in[23];
  const float* bn_g    = (const float*)d_in[24];
  const float* bn_b    = (const float*)d_in[25];
  const float* pred_W  = (const float*)d_in[26];
  const float* pred_b  = (const float*)d_in[27];

  const int n_low  = in_sizes[0] / (T_SEQ * G_HIN);
  const int n_high = in_sizes[1] / 6;
  const int e_l2h  = in_sizes[3];
  const int e_hh   = in_sizes[5];
  const int e_tot  = e_hh + n_high;

  // workspace layout (64-float aligned chunks)
  float* base = (float*)d_ws;
  size_t off = 0;
  auto alloc = [&](size_t n) { float* p = base + off; off += (n + 63) & ~(size_t)63; return p; };
  float* enc_raw = alloc((size_t)n_low * D_ENC);
  float* stat32  = alloc(64);
  float* s32     = alloc(32);
  float* t32     = alloc(32);
  float* low16   = alloc((size_t)n_low * D_HI);
  float* cntl    = alloc(n_high);
  float* cnth    = alloc(n_high);
  float* msg     = alloc((size_t)n_high * D_HI);
  float* ybufA   = alloc((size_t)n_high * D_HI);
  float* ybufB   = alloc((size_t)n_high * D_HI);
  float* xl      = alloc((size_t)n_high * D_HI);
  float* xr      = alloc((size_t)n_high * D_HI);
  float* ebuf    = alloc(e_tot);
  unsigned* emax = (unsigned*)alloc(n_high);
  float* denom   = alloc(n_high);
  float* stat16  = alloc(32);
  float* s16     = alloc(16);
  float* t16     = alloc(16);

  const int B = 256;
  const int nhi16 = n_high * D_HI;

  // --- encoder ---
  k_zero_f<<<1, 64, 0, stream>>>(stat32, 64);
  k_gru_dense<<<CDIV(n_low, 128), 128, 0, stream>>>(x_low, gru_Wih, gru_Whh, gru_bih, gru_bhh,
                                                    dense_W, dense_b, enc_raw, stat32, n_low);
  k_finalize_bn<<<1, 32, 0, stream>>>(stat32, bn_enc_g, bn_enc_b, s32, t32, 32, (float)n_low);

  // --- low->high projection (WMMA) + bipartite mean aggregation ---
  {
    int waves = CDIV(n_low, 16);
    k_down_project<<<CDIV(waves, 8), B, 0, stream>>>(enc_raw, s32, t32, down_Wrel, low16, n_low);
  }
  k_zero_f<<<CDIV(n_high, B), B, 0, stream>>>(cntl, n_high);
  k_zero_f<<<CDIV(n_high, B), B, 0, stream>>>(cnth, n_high);
  k_zero_f<<<CDIV(nhi16, B), B, 0, stream>>>(msg, nhi16);
  k_count<<<CDIV(e_l2h, B), B, 0, stream>>>(l2h_dst, cntl, e_l2h);
  k_count<<<CDIV(e_hh, B), B, 0, stream>>>(hh_dst, cnth, e_hh);
  k_scatter16<<<CDIV(e_l2h, B), B, 0, stream>>>(low16, l2h_src, l2h_dst, msg, e_l2h);
  k_zero_f<<<1, 32, 0, stream>>>(stat16, 32);
  k_high_init<<<CDIV(n_high, B), B, 0, stream>>>(msg, cntl, z_std, land, down_brel, down_Wroot,
                                                 ybufA, stat16, n_high);
  k_finalize_bn<<<1, 16, 0, stream>>>(stat16, bn_g, bn_b, s16, t16, 16, (float)n_high);  // BN0

  // --- 5 GATv2 layers ---
  float* yin = ybufA;
  float* yout = ybufB;
  int waves_hi = CDIV(n_high, 16);
  for (int i = 0; i < 5; i++) {
    k_gat_transform<<<CDIV(waves_hi, 8), B, 0, stream>>>(
        yin, s16, t16, /*apply_relu=*/(i > 0) ? 1 : 0,
        gat_Wl + (size_t)i * 256, gat_bl + (size_t)i * 16,
        gat_Wr + (size_t)i * 256, gat_br + (size_t)i * 16, xl, xr, n_high);
    k_zero_f<<<CDIV(n_high, B), B, 0, stream>>>((float*)emax, n_high);   // 0 == encoded -inf floor
    k_zero_f<<<CDIV(n_high, B), B, 0, stream>>>(denom, n_high);
    k_zero_f<<<CDIV(nhi16, B), B, 0, stream>>>(msg, nhi16);
    k_edge_logit<<<CDIV(e_tot, B), B, 0, stream>>>(xl, xr, hh_src, hh_dst,
                                                   gat_att + (size_t)i * 16, ebuf, emax, e_hh, e_tot);
    k_edge_exp<<<CDIV(e_tot, B), B, 0, stream>>>(hh_src, hh_dst, ebuf, emax, denom, e_hh, e_tot);
    k_edge_scatter<<<CDIV(e_tot, B), B, 0, stream>>>(xl, hh_src, hh_dst, ebuf, denom, msg, e_hh, e_tot);
    int collect = (i < 4) ? 1 : 0;
    if (collect) k_zero_f<<<1, 32, 0, stream>>>(stat16, 32);
    k_gat_finalize<<<CDIV(n_high, B), B, 0, stream>>>(msg, cnth, gat_bias + (size_t)i * 16,
                                                      yout, stat16, collect, n_high);
    if (collect)
      k_finalize_bn<<<1, 16, 0, stream>>>(stat16, bn_g + (size_t)(i + 1) * 16,
                                          bn_b + (size_t)(i + 1) * 16, s16, t16, 16, (float)n_high);
    float* tmp = yin; yin = yout; yout = tmp;
  }

  // --- predictor ---
  k_predict<<<CDIV(n_high, B), B, 0, stream>>>(yin, pred_W, pred_b, (float*)d_out, n_high);
}